// HypergraphRayleighQuotientLoss_60773787238819
// MI455X (gfx1250) — compile-verified
//
#include <hip/hip_runtime.h>
#include <hip/hip_bf16.h>
#include <math.h>

#define LAMBDA_ORTHO 0.1f

typedef __attribute__((ext_vector_type(2))) float v2f;
typedef __attribute__((ext_vector_type(8))) float v8f;

// ---------------------------------------------------------------------------
// Pass 1: node degree Dv (weight-weighted) and edge cardinality ne via atomics
// ---------------------------------------------------------------------------
__global__ void hg_degree_kernel(const int* __restrict__ node_idx,
                                 const int* __restrict__ edge_idx,
                                 const float* __restrict__ hw,
                                 float* __restrict__ Dv,
                                 float* __restrict__ ne,
                                 int nnz) {
    int t = blockIdx.x * blockDim.x + threadIdx.x;
    if (t >= nnz) return;
    int n = node_idx[t];
    int e = edge_idx[t];
    atomicAdd(&Dv[n], hw[e]);
    atomicAdd(&ne[e], 1.0f);
}

// ---------------------------------------------------------------------------
// Pass 2: Y = Z * Dv^{-1/2}  (Dv==0 clamped to 1)
// ---------------------------------------------------------------------------
__global__ void hg_scale_kernel(const float* __restrict__ Z,
                                const float* __restrict__ Dv,
                                float* __restrict__ Y,
                                int total /* N*64 */) {
    int tid = blockIdx.x * blockDim.x + threadIdx.x;
    if (tid >= total) return;
    int n = tid >> 6;
    float d = Dv[n];
    d = (d == 0.0f) ? 1.0f : d;
    Y[tid] = Z[tid] * rsqrtf(d);
}

// ---------------------------------------------------------------------------
// Pass 3: scatter per-edge vector sums S[e,:] += y, and accumulate the
// closed-form term1 = sum_t w_e * |y_node(t)|^2  (since n_e/De == 1 whenever
// an edge has members). 4 threads cooperate on one nnz (16 columns each).
// ---------------------------------------------------------------------------
__global__ __launch_bounds__(256) void hg_scatter_kernel(
        const float* __restrict__ Y,
        const int* __restrict__ node_idx,
        const int* __restrict__ edge_idx,
        const float* __restrict__ hw,
        float* __restrict__ S,
        float* __restrict__ acc,
        int nnz) {
    int tid = blockIdx.x * blockDim.x + threadIdx.x;
    int t = tid >> 2;
    float partial = 0.0f;
    if (t < nnz) {
        int c0 = (tid & 3) << 4;
        int n = node_idx[t];
        int e = edge_idx[t];
        const float4* yp = (const float4*)(Y + (size_t)n * 64 + c0);
        float* sp = S + (size_t)e * 64 + c0;
        float w = hw[e];
        float ss = 0.0f;
#pragma unroll
        for (int q = 0; q < 4; ++q) {
            float4 v = yp[q];
            atomicAdd(&sp[q * 4 + 0], v.x);
            atomicAdd(&sp[q * 4 + 1], v.y);
            atomicAdd(&sp[q * 4 + 2], v.z);
            atomicAdd(&sp[q * 4 + 3], v.w);
            ss += v.x * v.x + v.y * v.y + v.z * v.z + v.w * v.w;
        }
        partial = w * ss;
    }
    __shared__ float sdata[256];
    sdata[threadIdx.x] = partial;
    __syncthreads();
    for (int s = 128; s > 0; s >>= 1) {
        if (threadIdx.x < s) sdata[threadIdx.x] += sdata[threadIdx.x + s];
        __syncthreads();
    }
    if (threadIdx.x == 0) atomicAdd(acc, sdata[0]);
}

// ---------------------------------------------------------------------------
// Pass 4: Gram matrix G = Y^T Y (64x64) via V_WMMA_F32_16X16X4_F32.
// Block = 512 threads = 16 wave32s; wave w owns 16x16 tile (i0=(w/4)*16,
// j0=(w%4)*16). Each block reduces a slice of the K=N dimension; partial
// tiles accumulate into G with f32 atomics.
//
// f32 16x4 A layout: lanes 0-15 hold M=0..15 with VGPR0->K=0, VGPR1->K=1;
// lanes 16-31 hold VGPR0->K=2, VGPR1->K=3. The 4x16 B layout is the mirror
// (lane = N, lane-group selects K pair), so for a symmetric Gram product both
// operands are plain strided loads of Y rows.
//
// Unrolled 16 rows (4 WMMAs) per iteration from one moving base pointer so
// all 16 loads are immediate-offset b32s in a single clause, amortizing one
// s_wait_loadcnt over 4 WMMAs and killing per-iteration 64-bit address math.
// ---------------------------------------------------------------------------
__global__ __launch_bounds__(512) void hg_gram_wmma_kernel(
        const float* __restrict__ Y,
        float* __restrict__ G,
        int N, int rowsPerBlock) {
    int wave = threadIdx.x >> 5;
    int lane = threadIdx.x & 31;
    int i0 = (wave >> 2) << 4;
    int j0 = (wave & 3) << 4;
    int l16 = lane & 15;
    int hi = lane >> 4;

    int k = blockIdx.x * rowsPerBlock;
    int kend = min(k + rowsPerBlock, N);

    // Per-lane base pointers for row (k + 2*hi); a.y / b.y live one row (+64
    // floats) further.  All later accesses are compile-time byte offsets.
    const float* pA = Y + (size_t)(k + 2 * hi) * 64 + i0 + l16;
    const float* pB = Y + (size_t)(k + 2 * hi) * 64 + j0 + l16;

    v8f c = {};
    // Main loop: 16 K-rows, 4 WMMAs, 16 clause-able immediate-offset loads.
    for (; k + 16 <= kend; k += 16) {
        v2f a0, a1, a2, a3, b0, b1, b2, b3;
        a0.x = pA[0];   a0.y = pA[64];
        b0.x = pB[0];   b0.y = pB[64];
        a1.x = pA[256]; a1.y = pA[320];
        b1.x = pB[256]; b1.y = pB[320];
        a2.x = pA[512]; a2.y = pA[576];
        b2.x = pB[512]; b2.y = pB[576];
        a3.x = pA[768]; a3.y = pA[832];
        b3.x = pB[768]; b3.y = pB[832];
        c = __builtin_amdgcn_wmma_f32_16x16x4_f32(false, a0, false, b0, (short)0, c, false, false);
        c = __builtin_amdgcn_wmma_f32_16x16x4_f32(false, a1, false, b1, (short)0, c, false, false);
        c = __builtin_amdgcn_wmma_f32_16x16x4_f32(false, a2, false, b2, (short)0, c, false, false);
        c = __builtin_amdgcn_wmma_f32_16x16x4_f32(false, a3, false, b3, (short)0, c, false, false);
        pA += 1024;
        pB += 1024;
    }
    // 4-row steps for the remainder.
    for (; k + 4 <= kend; k += 4) {
        v2f a, b;
        a.x = pA[0]; a.y = pA[64];
        b.x = pB[0]; b.y = pB[64];
        c = __builtin_amdgcn_wmma_f32_16x16x4_f32(false, a, false, b, (short)0, c, false, false);
        pA += 256;
        pB += 256;
    }
    if (k < kend) {  // <4 rows left (uniform k per wave; guards reconverge)
        int ka = k + 2 * hi;
        v2f a = {}, b = {};
        if (ka < kend)     { a.x = pA[0];  b.x = pB[0]; }
        if (ka + 1 < kend) { a.y = pA[64]; b.y = pB[64]; }
        c = __builtin_amdgcn_wmma_f32_16x16x4_f32(false, a, false, b, (short)0, c, false, false);
    }

    // C/D layout: VGPR r, lanes 0-15 -> (M=r, N=lane); lanes 16-31 -> (M=r+8).
#pragma unroll
    for (int r = 0; r < 8; ++r) {
        int M = r + 8 * hi;
        atomicAdd(&G[(size_t)(i0 + M) * 64 + (j0 + l16)], c[r]);
    }
}

// ---------------------------------------------------------------------------
// Pass 5: per-edge  -= (w_e / De) * |S_e|^2
// ---------------------------------------------------------------------------
__global__ __launch_bounds__(256) void hg_edge_finalize_kernel(
        const float* __restrict__ S,
        const float* __restrict__ ne,
        const float* __restrict__ hw,
        float* __restrict__ acc,
        int E) {
    int e = blockIdx.x * blockDim.x + threadIdx.x;
    float partial = 0.0f;
    if (e < E) {
        float cnt = ne[e];
        float De = (cnt == 0.0f) ? 1.0f : cnt;
        const float4* sp = (const float4*)(S + (size_t)e * 64);
        float s2 = 0.0f;
#pragma unroll
        for (int q = 0; q < 16; ++q) {
            float4 v = sp[q];
            s2 += v.x * v.x + v.y * v.y + v.z * v.z + v.w * v.w;
        }
        partial = -(hw[e] / De) * s2;
    }
    __shared__ float sdata[256];
    sdata[threadIdx.x] = partial;
    __syncthreads();
    for (int s = 128; s > 0; s >>= 1) {
        if (threadIdx.x < s) sdata[threadIdx.x] += sdata[threadIdx.x + s];
        __syncthreads();
    }
    if (threadIdx.x == 0) atomicAdd(acc, sdata[0]);
}

// ---------------------------------------------------------------------------
// Pass 6: out = acc + LAMBDA * ||G - I||_F   (single block)
// ---------------------------------------------------------------------------
__global__ __launch_bounds__(256) void hg_final_kernel(
        const float* __restrict__ G,
        const float* __restrict__ acc,
        float* __restrict__ out) {
    float s = 0.0f;
    for (int i = threadIdx.x; i < 64 * 64; i += 256) {
        int row = i >> 6, col = i & 63;
        float d = G[i] - ((row == col) ? 1.0f : 0.0f);
        s += d * d;
    }
    __shared__ float sdata[256];
    sdata[threadIdx.x] = s;
    __syncthreads();
    for (int st = 128; st > 0; st >>= 1) {
        if (threadIdx.x < st) sdata[threadIdx.x] += sdata[threadIdx.x + st];
        __syncthreads();
    }
    if (threadIdx.x == 0) out[0] = acc[0] + LAMBDA_ORTHO * sqrtf(sdata[0]);
}

// ---------------------------------------------------------------------------
static inline size_t align4f(size_t x) { return (x + 3) & ~(size_t)3; }

extern "C" void kernel_launch(void* const* d_in, const int* in_sizes, int n_in,
                              void* d_out, int out_size, void* d_ws, size_t ws_size,
                              hipStream_t stream) {
    const float* Z  = (const float*)d_in[0];
    const int*   hi = (const int*)d_in[1];
    const float* hw = (const float*)d_in[3];

    const int NK  = in_sizes[0];          // N * 64
    const int N   = NK >> 6;
    const int nnz = in_sizes[1] >> 1;     // hyperedge_index is [2, nnz]
    const int E   = in_sizes[3];

    const int* node_idx = hi;
    const int* edge_idx = hi + nnz;

    // Workspace layout (floats), all 16B-aligned chunks:
    float* W = (float*)d_ws;
    size_t o_dv  = 0;
    size_t o_ne  = o_dv + align4f((size_t)N);
    size_t o_s   = o_ne + align4f((size_t)E);
    size_t o_g   = o_s  + align4f((size_t)E * 64);
    size_t o_acc = o_g  + 64 * 64;
    size_t o_y   = align4f(o_acc + 4);

    float* Dv  = W + o_dv;
    float* ne  = W + o_ne;
    float* S   = W + o_s;
    float* G   = W + o_g;
    float* acc = W + o_acc;
    float* Y   = W + o_y;

    // Zero all accumulators (Dv, ne, S, G, acc) in one async memset.
    hipMemsetAsync(d_ws, 0, o_y * sizeof(float), stream);

    // 1) degrees + cardinalities
    hg_degree_kernel<<<(nnz + 255) / 256, 256, 0, stream>>>(
        node_idx, edge_idx, hw, Dv, ne, nnz);

    // 2) Y = Z * Dv^{-1/2}
    hg_scale_kernel<<<(NK + 255) / 256, 256, 0, stream>>>(Z, Dv, Y, NK);

    // 3) scatter S[e,:] += y ; acc += w_e * |y|^2
    {
        long long threads = (long long)nnz * 4;
        hg_scatter_kernel<<<(int)((threads + 255) / 256), 256, 0, stream>>>(
            Y, node_idx, edge_idx, hw, S, acc, nnz);
    }

    // 4) Gram matrix via WMMA f32 16x16x4
    {
        const int rowsPerBlock = 4096;
        int nblocks = (N + rowsPerBlock - 1) / rowsPerBlock;
        hg_gram_wmma_kernel<<<nblocks, 512, 0, stream>>>(Y, G, N, rowsPerBlock);
    }

    // 5) acc -= (w_e/De) * |S_e|^2
    hg_edge_finalize_kernel<<<(E + 255) / 256, 256, 0, stream>>>(S, ne, hw, acc, E);

    // 6) out = acc + 0.1 * ||G - I||_F
    hg_final_kernel<<<1, 256, 0, stream>>>(G, acc, (float*)d_out);
}